// Gaussian_Distance_KL_19172734009624
// MI455X (gfx1250) — compile-verified
//
#include <hip/hip_runtime.h>
#include <math.h>

typedef __attribute__((ext_vector_type(2))) float v2f;
typedef __attribute__((ext_vector_type(8))) float v8f;

#define EPS_F 1e-6f

__global__ void gdkl_zero_out(float* out) { out[0] = 0.f; }

// ds_swizzle_b32 group-of-32 XOR pattern: offset = {xor[14:10], or[9:5]=0, and[4:0]=0x1f}
template<int M>
__device__ __forceinline__ float swz_xor(float x) {
  return __int_as_float(
      __builtin_amdgcn_ds_swizzle(__float_as_int(x), (M << 10) | 0x1f));
}

// Merge two row-registers while folding lane-distance M: lanes with bit M==0
// end up holding row-a pair sums, lanes with bit M==1 hold row-b pair sums.
template<int M>
__device__ __forceinline__ float merge_xor(float a, float b, int lane) {
  float keep = (lane & M) ? b : a;
  float send = (lane & M) ? a : b;
  return keep + swz_xor<M>(send);
}

template<int M>
__device__ __forceinline__ float bfly_xor(float x) { return x + swz_xor<M>(x); }

// B=32, C=128, H=W=32, kern=2 -> h=w=16. One block per (c,h).
__launch_bounds__(256)
__global__ void gdkl_kernel(const float* __restrict__ mu_a,
                            const float* __restrict__ logvar_a,
                            const float* __restrict__ mu_b,
                            const float* __restrict__ logvar_b,
                            const float* __restrict__ eps,
                            float* __restrict__ out) {
  // Per-(i,w) precomputed values, indexed [w*32 + i] (i contiguous -> conflict-free
  // LDS reads by lanes in stage 2).
  __shared__ float s_z [512];
  __shared__ float s_a0[512], s_a1[512], s_a2[512];   // density-a coeffs (log2 domain)
  __shared__ float s_b0[512], s_b1[512], s_b2[512];   // density-b coeffs (log2 domain)

  const int c = blockIdx.x >> 4;    // 0..127
  const int h = blockIdx.x & 15;    // 0..15
  const int t = threadIdx.x;
  const float LOG2E = 1.4426950408889634f;

  // ---- Stage 1: fused avg-pool + reparameterization + exponent coefficients ----
  for (int idx = t; idx < 512; idx += 256) {
    const int i = idx >> 4;         // batch row
    const int w = idx & 15;         // pooled column
    const size_t base = (((size_t)i * 128 + c) * 32 + 2 * h) * 32 + 2 * w;

    float2 m0 = *(const float2*)(mu_a + base);
    float2 m1 = *(const float2*)(mu_a + base + 32);
    float ma = (m0.x + m0.y + m1.x + m1.y) * 0.25f;

    float2 la0 = *(const float2*)(logvar_a + base);
    float2 la1 = *(const float2*)(logvar_a + base + 32);
    float va = (__builtin_amdgcn_exp2f(la0.x * LOG2E) + __builtin_amdgcn_exp2f(la0.y * LOG2E) +
                __builtin_amdgcn_exp2f(la1.x * LOG2E) + __builtin_amdgcn_exp2f(la1.y * LOG2E)) * (1.f / 16.f);

    float2 n0 = *(const float2*)(mu_b + base);
    float2 n1 = *(const float2*)(mu_b + base + 32);
    float mb = (n0.x + n0.y + n1.x + n1.y) * 0.25f;

    float2 lb0 = *(const float2*)(logvar_b + base);
    float2 lb1 = *(const float2*)(logvar_b + base + 32);
    float vb = (__builtin_amdgcn_exp2f(lb0.x * LOG2E) + __builtin_amdgcn_exp2f(lb0.y * LOG2E) +
                __builtin_amdgcn_exp2f(lb1.x * LOG2E) + __builtin_amdgcn_exp2f(lb1.y * LOG2E)) * (1.f / 16.f);

    float e = eps[(((size_t)i * 128 + c) * 16 + h) * 16 + w];
    float z = ma + __builtin_amdgcn_sqrtf(va) * e;

    const int o = w * 32 + i;
    s_z[o] = z;
    // log2(density) = z^2*(-log2e/2v) + z*(mu*log2e/v) + (-mu^2*log2e/2v - 0.5*log2 v)
    float ra = __builtin_amdgcn_rcpf(va);
    s_a0[o] = -0.5f * ra * LOG2E;
    s_a1[o] = ma * ra * LOG2E;
    s_a2[o] = -0.5f * ma * ma * ra * LOG2E - 0.5f * __builtin_amdgcn_logf(va);
    float rb = __builtin_amdgcn_rcpf(vb);
    s_b0[o] = -0.5f * rb * LOG2E;
    s_b1[o] = mb * rb * LOG2E;
    s_b2[o] = -0.5f * mb * mb * rb * LOG2E - 0.5f * __builtin_amdgcn_logf(vb);
  }
  __syncthreads();

  // ---- Stage 2: 32x32 pairwise exponent matrices via WMMA f32 16x16x4 ----
  const int wv   = t >> 5;          // wave id 0..7
  const int lane = t & 31;
  const int l16  = lane & 15;
  const bool lo  = lane < 16;

  float acc = 0.f;                  // accumulates +/- log2(mean + EPS)
  for (int which = 0; which < 2; ++which) {
    const int w  = wv + which * 8;  // this wave's pooled column
    const int wb = w * 32;

    // A tiles (16x4, rows i, cols [z^2, z, 1, 0]):
    // lanes 0-15 hold K0/K1 (z^2, z); lanes 16-31 hold K2/K3 (1, 0).
    float z0 = s_z[wb + l16];         // rows 0..15
    float z1 = s_z[wb + 16 + l16];    // rows 16..31
    v2f a0f, a1f;
    a0f.x = lo ? z0 * z0 : 1.f;  a0f.y = lo ? z0 : 0.f;
    a1f.x = lo ? z1 * z1 : 1.f;  a1f.y = lo ? z1 : 0.f;

    for (int den = 0; den < 2; ++den) {
      const float* c0 = den ? s_b0 : s_a0;
      const float* c1 = den ? s_b1 : s_a1;
      const float* c2 = den ? s_b2 : s_a2;

      // B tiles (4x16, rows K, cols j): lanes 0-15 hold K0/K1 rows,
      // lanes 16-31 hold K2/K3 rows, mirroring the A layout.
      float p0 = c0[wb + l16],      q0 = c1[wb + l16],      r0 = c2[wb + l16];
      float p1 = c0[wb + 16 + l16], q1 = c1[wb + 16 + l16], r1 = c2[wb + 16 + l16];
      v2f b0f, b1f;
      b0f.x = lo ? p0 : r0;  b0f.y = lo ? q0 : 0.f;
      b1f.x = lo ? p1 : r1;  b1f.y = lo ? q1 : 0.f;

      v8f zc = {};
      v8f d00 = __builtin_amdgcn_wmma_f32_16x16x4_f32(false, a0f, false, b0f, (short)0, zc, false, false);
      v8f d01 = __builtin_amdgcn_wmma_f32_16x16x4_f32(false, a0f, false, b1f, (short)0, zc, false, false);
      v8f d10 = __builtin_amdgcn_wmma_f32_16x16x4_f32(false, a1f, false, b0f, (short)0, zc, false, false);
      v8f d11 = __builtin_amdgcn_wmma_f32_16x16x4_f32(false, a1f, false, b1f, (short)0, zc, false, false);

      // Raw v_exp_f32 per pair (exponents already in log2 domain).
      // r[0..7]  = rows  v /  v+8 (low/high lane halves), summed over j-tiles
      // r[8..15] = rows 16+v / 24+v
      float r[16];
      #pragma unroll
      for (int v = 0; v < 8; ++v) {
        r[v]     = __builtin_amdgcn_exp2f(d00[v]) + __builtin_amdgcn_exp2f(d01[v]);
        r[8 + v] = __builtin_amdgcn_exp2f(d10[v]) + __builtin_amdgcn_exp2f(d11[v]);
      }

      // Merging reduction tree on immediate-pattern ds_swizzle: 16 regs x 16
      // lanes (per half) -> 1 reg where every one of the 32 lanes holds
      // exactly one full row-sum (all 32 rows once across the wave).
      float s0 = merge_xor<1>(r[0],  r[1],  lane);
      float s1 = merge_xor<1>(r[2],  r[3],  lane);
      float s2 = merge_xor<1>(r[4],  r[5],  lane);
      float s3 = merge_xor<1>(r[6],  r[7],  lane);
      float s4 = merge_xor<1>(r[8],  r[9],  lane);
      float s5 = merge_xor<1>(r[10], r[11], lane);
      float s6 = merge_xor<1>(r[12], r[13], lane);
      float s7 = merge_xor<1>(r[14], r[15], lane);

      float t0 = merge_xor<2>(s0, s1, lane);
      float t1 = merge_xor<2>(s2, s3, lane);
      float t2 = merge_xor<2>(s4, s5, lane);
      float t3 = merge_xor<2>(s6, s7, lane);

      float u0 = merge_xor<4>(t0, t1, lane);
      float u1 = merge_xor<4>(t2, t3, lane);

      float rowsum = merge_xor<8>(u0, u1, lane);

      const float sign = den ? -1.f : 1.f;
      acc += sign * __builtin_amdgcn_logf(rowsum * (1.f / 32.f) + EPS_F);
    }
  }

  // acc holds log2-domain terms, each (i,c,h,w) row term counted exactly once.
  // reference = B * ln2 * sum(log2 terms);  B = 32.
  acc = bfly_xor<1>(acc);
  acc = bfly_xor<2>(acc);
  acc = bfly_xor<4>(acc);
  acc = bfly_xor<8>(acc);
  acc = bfly_xor<16>(acc);
  if (lane == 0) atomicAdd(out, acc * (32.0f * 0.69314718055994531f));
}

extern "C" void kernel_launch(void* const* d_in, const int* in_sizes, int n_in,
                              void* d_out, int out_size, void* d_ws, size_t ws_size,
                              hipStream_t stream) {
  const float* mu_a = (const float*)d_in[0];
  const float* lva  = (const float*)d_in[1];
  const float* mu_b = (const float*)d_in[2];
  const float* lvb  = (const float*)d_in[3];
  const float* eps  = (const float*)d_in[4];
  float* out = (float*)d_out;

  gdkl_zero_out<<<1, 1, 0, stream>>>(out);
  gdkl_kernel<<<2048, 256, 0, stream>>>(mu_a, lva, mu_b, lvb, eps, out);
}